// Alignment_8899172238147
// MI455X (gfx1250) — compile-verified
//
#include <hip/hip_runtime.h>
#include <math.h>

// CDNA5 / gfx1250, wave32.
// v_wmma_f32_16x16x4_f32: A 16x4 f32 (2 VGPR), B 4x16 f32 (2 VGPR), C/D 16x16 f32 (8 VGPR).
// A layout (ISA 7.12.2): lanes 0-15 = M, VGPR0 holds K=0 (lanes0-15)/K=2 (lanes16-31),
// VGPR1 holds K=1/K=3. B layout is assumed mirror-symmetric (N across lanes, same K slots).
// Block-diagonal packing: row m = (batch q=m/3, comp i=m%3), col n = (batch q=n/3, comp j=n%3);
// the five diagonal 3x3 blocks of the accumulator are five batches' H = fx^T fy.

typedef float v2f __attribute__((ext_vector_type(2)));
typedef float v8f __attribute__((ext_vector_type(8)));

#define WAVES_PB 8
#define BATCH_PW 5
#define BATCH_PB (WAVES_PB * BATCH_PW) /* 40 */

__global__ __launch_bounds__(256)
void kabsch_wmma_kernel(const float* __restrict__ fx, const float* __restrict__ fy,
                        float* __restrict__ out, int B, int C)
{
    __shared__ float sC[WAVES_PB][8][32];   // 8 KB: 16x16 f32 accumulator per wave
    __shared__ float sNa[WAVES_PB][32];     // per-lane sum fx^2
    __shared__ float sNb[WAVES_PB][32];     // per-lane sum fy^2

    const int lane = threadIdx.x & 31;
    const int wv   = threadIdx.x >> 5;
    const int half = lane >> 4;     // 0: K slots {0,1}; 1: K slots {2,3}
    const int mn   = lane & 15;     // M (A) / N (B) index on the 16x16 tile
    int q    = mn / 3;              // batch-in-wave 0..4 (mn==15 unused)
    int comp = mn - q * 3;          // component 0..2
    if (mn == 15) { q = 0; comp = 0; }              // clamp dead lane to safe addresses
    long b = (long)(blockIdx.x * WAVES_PB + wv) * BATCH_PW + q;
    if (b >= B) b = 0;                               // tail clamp; stores are guarded below

    const size_t base = (size_t)b * (size_t)C * 3u + (size_t)(half * 2) * 3u + (size_t)comp;
    const float* pa = fx + base;
    const float* pb = fy + base;

    v8f acc = {0.f, 0.f, 0.f, 0.f, 0.f, 0.f, 0.f, 0.f};
    float na = 0.f, nb = 0.f;

    // 4 points per WMMA, unroll x4 => 16 points per trip, C/16 trips.
    for (int c0 = 0; c0 < C; c0 += 16) {
        #pragma unroll
        for (int j = 0; j < 4; ++j) {
            v2f av, bv;
            av.x = pa[j * 12 + 0];   // K slot (half*2)
            av.y = pa[j * 12 + 3];   // K slot (half*2 + 1)
            bv.x = pb[j * 12 + 0];
            bv.y = pb[j * 12 + 3];
            acc = __builtin_amdgcn_wmma_f32_16x16x4_f32(false, av, false, bv,
                                                        (short)0, acc, false, false);
            na = fmaf(av.x, av.x, na); na = fmaf(av.y, av.y, na);
            nb = fmaf(bv.x, bv.x, nb); nb = fmaf(bv.y, bv.y, nb);
        }
        pa += 48; pb += 48;
    }

    #pragma unroll
    for (int r = 0; r < 8; ++r) sC[wv][r][lane] = acc[r];
    sNa[wv][lane] = na;
    sNb[wv][lane] = nb;
    __syncthreads();

    // ---- epilogue: one thread per batch does the 3x3 SVD + Kabsch ----
    const int t = threadIdx.x;
    if (t < BATCH_PB) {
        const int w2 = t / BATCH_PW;
        const int q2 = t - w2 * BATCH_PW;
        const long bo = (long)(blockIdx.x * WAVES_PB + w2) * BATCH_PW + q2;
        if (bo < B) {
            // Extract H from the C/D tile layout: VGPR r, lanes0-15 => M=r; lanes16-31 => M=r+8.
            float Hm[3][3];
            #pragma unroll
            for (int i = 0; i < 3; ++i) {
                #pragma unroll
                for (int j = 0; j < 3; ++j) {
                    const int M = 3 * q2 + i, N = 3 * q2 + j;
                    Hm[i][j] = sC[w2][M & 7][(M < 8) ? N : (N + 16)];
                }
            }

            // A = H^T H (symmetric PSD), Jacobi eigendecomposition -> V, lambda.
            float Am[3][3];
            #pragma unroll
            for (int i = 0; i < 3; ++i)
                #pragma unroll
                for (int j = 0; j < 3; ++j)
                    Am[i][j] = Hm[0][i]*Hm[0][j] + Hm[1][i]*Hm[1][j] + Hm[2][i]*Hm[2][j];

            float Vm[3][3] = {{1.f,0.f,0.f},{0.f,1.f,0.f},{0.f,0.f,1.f}};

            #define JROT(p, qq) do {                                                  \
                float apq = Am[p][qq];                                                \
                if (fabsf(apq) > 1e-18f) {                                            \
                    float tau = (Am[qq][qq] - Am[p][p]) / (2.0f * apq);               \
                    float tt  = copysignf(1.0f, tau) /                                \
                                (fabsf(tau) + sqrtf(fmaf(tau, tau, 1.0f)));           \
                    float cc  = rsqrtf(fmaf(tt, tt, 1.0f));                           \
                    float ss  = tt * cc;                                              \
                    _Pragma("unroll")                                                 \
                    for (int k = 0; k < 3; ++k) {                                     \
                        float akp = Am[k][p], akq = Am[k][qq];                        \
                        Am[k][p]  = cc*akp - ss*akq;                                  \
                        Am[k][qq] = ss*akp + cc*akq;                                  \
                    }                                                                 \
                    _Pragma("unroll")                                                 \
                    for (int k = 0; k < 3; ++k) {                                     \
                        float apk = Am[p][k], aqk = Am[qq][k];                        \
                        Am[p][k]  = cc*apk - ss*aqk;                                  \
                        Am[qq][k] = ss*apk + cc*aqk;                                  \
                    }                                                                 \
                    _Pragma("unroll")                                                 \
                    for (int k = 0; k < 3; ++k) {                                     \
                        float vkp = Vm[k][p], vkq = Vm[k][qq];                        \
                        Vm[k][p]  = cc*vkp - ss*vkq;                                  \
                        Vm[k][qq] = ss*vkp + cc*vkq;                                  \
                    }                                                                 \
                }                                                                     \
            } while (0)

            #pragma unroll
            for (int sweep = 0; sweep < 6; ++sweep) {
                JROT(0, 1); JROT(0, 2); JROT(1, 2);
            }
            #undef JROT

            // U_j = H v_j / sigma_j ; R = V U^T  (sign/order invariant, matches v @ u.T)
            float R[3][3] = {{0.f,0.f,0.f},{0.f,0.f,0.f},{0.f,0.f,0.f}};
            #pragma unroll
            for (int j = 0; j < 3; ++j) {
                float sig = sqrtf(fmaxf(Am[j][j], 0.0f));
                float inv = 1.0f / fmaxf(sig, 1e-20f);
                float u0 = (Hm[0][0]*Vm[0][j] + Hm[0][1]*Vm[1][j] + Hm[0][2]*Vm[2][j]) * inv;
                float u1 = (Hm[1][0]*Vm[0][j] + Hm[1][1]*Vm[1][j] + Hm[1][2]*Vm[2][j]) * inv;
                float u2 = (Hm[2][0]*Vm[0][j] + Hm[2][1]*Vm[1][j] + Hm[2][2]*Vm[2][j]) * inv;
                #pragma unroll
                for (int i = 0; i < 3; ++i) {
                    R[i][0] = fmaf(Vm[i][j], u0, R[i][0]);
                    R[i][1] = fmaf(Vm[i][j], u1, R[i][1]);
                    R[i][2] = fmaf(Vm[i][j], u2, R[i][2]);
                }
            }

            float* Ro = out + (size_t)bo * 9u;
            #pragma unroll
            for (int i = 0; i < 3; ++i)
                #pragma unroll
                for (int j = 0; j < 3; ++j)
                    Ro[i * 3 + j] = R[i][j];

            // S[i] = ||fy[:,i]|| / ||fx[:,i]||  (combine the two K-half lane partials)
            float* So = out + (size_t)B * 9u + (size_t)bo * 3u;
            #pragma unroll
            for (int i = 0; i < 3; ++i) {
                const int li = 3 * q2 + i;
                float nfx = sNa[w2][li] + sNa[w2][16 + li];
                float nfy = sNb[w2][li] + sNb[w2][16 + li];
                So[i] = sqrtf(nfy / fmaxf(nfx, 1e-30f));
            }
        }
    }
}

extern "C" void kernel_launch(void* const* d_in, const int* in_sizes, int n_in,
                              void* d_out, int out_size, void* d_ws, size_t ws_size,
                              hipStream_t stream) {
    (void)n_in; (void)out_size; (void)d_ws; (void)ws_size;
    const float* fx = (const float*)d_in[0];
    const float* fy = (const float*)d_in[1];
    float* out = (float*)d_out;

    const int C = 1024;                       // points per batch (reference-fixed)
    const int B = in_sizes[0] / (C * 3);      // 16384
    const int nblk = (B + BATCH_PB - 1) / BATCH_PB;
    kabsch_wmma_kernel<<<nblk, 256, 0, stream>>>(fx, fy, out, B, C);
}